// FusedLocalNormalizedCrossCorrelationLoss_88441966559925
// MI455X (gfx1250) — compile-verified
//
#include <hip/hip_runtime.h>

// Fused Local NCC loss, 3x3x3 window, 192^3 f32 -> scalar. Bandwidth-bound
// (~57 MB reads => ~2.4us @ 23.3 TB/s HBM). One wave32 per 14x14 (x,y) tile
// per 24-plane z-chunk:
//   - TDM (tensor_load_to_lds) double-buffers 16x16 plane tiles of pred/targ
//     into LDS, with D# padding giving a 17-dword LDS row stride and D#
//     tensor_dim OOB producing the zero "same" padding at right/bottom edges.
//   - 2D 3x3 box sums of the 5 products computed as T*(M*T), T = 16x16
//     tridiagonal ones matrix, via chained V_WMMA_F32_16X16X4_F32; the
//     clipped band at matrix edges implements zero-padding at left/top.
//   - z-direction 3-sum via register ring; per-voxel NCC in VALU; block
//     partials reduced by a tiny deterministic second kernel.

typedef __attribute__((ext_vector_type(2))) float        v2f;
typedef __attribute__((ext_vector_type(8))) float        v8f;
typedef __attribute__((ext_vector_type(4))) unsigned int v4u;
typedef __attribute__((ext_vector_type(4))) int          v4i;
typedef __attribute__((ext_vector_type(8))) int          v8i;

#define N      192
#define OT     14
#define NT     14            // ceil(192/14)
#define CHUNK  24
#define ZCH    8             // 192/24
#define LST    17            // LDS row stride in dwords (17 = 16 + TDM pad)
#define TILE   (16 * LST)    // dwords per 16x16 tile in LDS
#define NBLK   (NT * NT * ZCH)

#if __has_builtin(__builtin_amdgcn_tensor_load_to_lds)
#define HAVE_TDM 1
#else
#define HAVE_TDM 0
#endif

__device__ __forceinline__ float band1(int a, int b) {
    int d = a - b;
    return (d >= -1 && d <= 1) ? 1.0f : 0.0f;
}

__device__ __forceinline__ void wait_tensorcnt(bool two) {
#if HAVE_TDM
#if __has_builtin(__builtin_amdgcn_s_wait_tensorcnt)
    if (two) __builtin_amdgcn_s_wait_tensorcnt(2);
    else     __builtin_amdgcn_s_wait_tensorcnt(0);
#else
    if (two) asm volatile("s_wait_tensorcnt 0x2" ::: "memory");
    else     asm volatile("s_wait_tensorcnt 0x0" ::: "memory");
#endif
#endif
    asm volatile("" ::: "memory");
}

// D(16x16) = A(16x4) x B(4x16) + C, f32, wave32 (layouts per CDNA5 ISA).
__device__ __forceinline__ v8f wmma_16x16x4_f32(v2f a, v2f b, v8f c) {
#if __has_builtin(__builtin_amdgcn_wmma_f32_16x16x4_f32)
    return __builtin_amdgcn_wmma_f32_16x16x4_f32(false, a, false, b,
                                                 (short)0, c, false, false);
#else
    const int lane = threadIdx.x & 31;
    const int half = lane >> 4, l16 = lane & 15;
    v8f d;
#pragma unroll
    for (int r = 0; r < 8; ++r) {
        int M = r + 8 * half;
        float s = c[r];
#pragma unroll
        for (int k = 0; k < 4; ++k) {
            float av = __shfl(a[k & 1], M   + 16 * (k >> 1), 32);
            float bv = __shfl(b[k & 1], l16 + 16 * (k >> 1), 32);
            s += av * bv;
        }
        d[r] = s;
    }
    return d;
#endif
}

__global__ __launch_bounds__(32)
void lncc_main(const float* __restrict__ pred,
               const float* __restrict__ targ,
               float* __restrict__ partial)
{
    // [buf0: sp,st][buf1: sp,st][sscr]
    __shared__ float smem[5 * TILE];

    const int lane = threadIdx.x;          // single wave32 per block
    const int bid  = blockIdx.x;
    const int tx   = bid % NT;
    const int ty   = (bid / NT) % NT;
    const int z0   = (bid / (NT * NT)) * CHUNK;
    const int half = lane >> 4, l16 = lane & 15;

    // Edge tiles start at 0 and rely on the clipped band = zero padding.
    const int x0  = (tx == 0) ? 0 : (tx * OT - 1);
    const int y0  = (ty == 0) ? 0 : (ty * OT - 1);
    const int lox = (tx == 0) ? 0 : 1;
    const int hix = (tx == 0) ? 13 : 14;
    const int loy = (ty == 0) ? 0 : 1;
    const int hiy = (ty == 0) ? 13 : 14;

    // Tridiagonal-ones operand fragments (symmetric: serves A' and B).
    float Tf[4][2];
#pragma unroll
    for (int c = 0; c < 4; ++c)
#pragma unroll
        for (int v = 0; v < 2; ++v)
            Tf[c][v] = band1(l16, 4 * c + v + 2 * half);

    const unsigned lds_base = (unsigned)(size_t)(void*)smem;

    // Stage plane zp into buffer buf. Returns true iff TDM ops were issued.
    auto stage = [&](int zp, int buf) -> bool {
        const int cb = buf * 2 * TILE;
        if ((unsigned)zp < (unsigned)N) {
#if HAVE_TDM
            unsigned long long off =
                (unsigned long long)(((size_t)zp * N + y0) * N + x0) * 4ull;
            unsigned long long pa = (unsigned long long)(size_t)pred + off;
            unsigned long long ta = (unsigned long long)(size_t)targ + off;
            unsigned td0 = (unsigned)(N - x0);   // right-edge OOB -> zeros
            unsigned td1 = (unsigned)(N - y0);   // bottom-edge OOB -> zeros
            v8i g1;
            g1[0] = (2 << 16) | (1 << 20) | (3 << 22); // 4B, pad 1dw/16dw
            g1[1] = (int)((td0 & 0xFFFFu) << 16);
            g1[2] = (int)((td0 >> 16) | ((td1 & 0xFFFFu) << 16));
            g1[3] = (int)((td1 >> 16) | (16u << 16));  // tile_dim0 = 16
            g1[4] = 16;                                // tile_dim1=16, dim2=0
            g1[5] = N;                                 // dim0 stride
            g1[6] = (int)(((unsigned)(N * N) & 0xFFFFu) << 16); // dim1 stride lo
            g1[7] = (int)((unsigned)(N * N) >> 16);             // dim1 stride hi
            v4i gz4 = {0, 0, 0, 0};
            v8i gz8 = {0, 0, 0, 0, 0, 0, 0, 0};
            unsigned cbb = (unsigned)cb * 4u;
            v4u g0a = {1u, lds_base + cbb,
                       (unsigned)pa, (unsigned)(pa >> 32) | (2u << 30)};
            __builtin_amdgcn_tensor_load_to_lds(g0a, g1, gz4, gz4, gz8, 0);
            v4u g0b = {1u, lds_base + cbb + (unsigned)TILE * 4u,
                       (unsigned)ta, (unsigned)(ta >> 32) | (2u << 30)};
            __builtin_amdgcn_tensor_load_to_lds(g0b, g1, gz4, gz4, gz8, 0);
            return true;
#else
#pragma unroll
            for (int j = 0; j < 8; ++j) {
                int i = lane + 32 * j;
                int row = i >> 4, col = i & 15;
                int gx = x0 + col, gy = y0 + row;
                bool ok = (gx < N) && (gy < N);
                size_t idx = ((size_t)zp * N + gy) * N + gx;
                smem[cb + row * LST + col]        = ok ? pred[idx] : 0.0f;
                smem[cb + TILE + row * LST + col] = ok ? targ[idx] : 0.0f;
            }
            return false;
#endif
        } else {
#pragma unroll
            for (int j = 0; j < 8; ++j) {
                int i = lane + 32 * j;
                int row = i >> 4, col = i & 15;
                smem[cb + row * LST + col]        = 0.0f;
                smem[cb + TILE + row * LST + col] = 0.0f;
            }
            return false;
        }
    };

    v8f Pa[5], Pb[5], Pc[5];
#pragma unroll
    for (int p = 0; p < 5; ++p) {
        Pa[p] = (v8f){0,0,0,0,0,0,0,0};
        Pb[p] = (v8f){0,0,0,0,0,0,0,0};
        Pc[p] = (v8f){0,0,0,0,0,0,0,0};
    }
    float acc = 0.0f;

    const int zlo = z0 - 1, zhi = z0 + CHUNK;
    stage(zlo, 0);

    for (int zp = zlo; zp <= zhi; ++zp) {
        const int cur = (zp - zlo) & 1;
        const int cb  = cur * 2 * TILE;

        // All DS reads of the buffer we are about to refill must retire
        // first: TENSOR and DS ops are mutually unordered.
        asm volatile("s_wait_dscnt 0x0" ::: "memory");
        bool ntdm = false;
        if (zp < zhi) ntdm = stage(zp + 1, cur ^ 1);
        wait_tensorcnt(ntdm);   // <=2 leaves only the next plane's pair in flight

        // ---- P(zp) = T * (M_p * T): 2D 3x3 box sums of the 5 products ----
        v8f Pnew[5];
#pragma unroll
        for (int p = 0; p < 5; ++p) {
            v8f d = (v8f){0,0,0,0,0,0,0,0};
#pragma unroll
            for (int c = 0; c < 4; ++c) {
                int col0 = 4 * c + 2 * half;
                float p0 = smem[cb + l16 * LST + col0];
                float p1 = smem[cb + l16 * LST + col0 + 1];
                float t0 = smem[cb + TILE + l16 * LST + col0];
                float t1 = smem[cb + TILE + l16 * LST + col0 + 1];
                float a0, a1;
                if      (p == 0) { a0 = p0;      a1 = p1;      }
                else if (p == 1) { a0 = t0;      a1 = t1;      }
                else if (p == 2) { a0 = p0 * p0; a1 = p1 * p1; }
                else if (p == 3) { a0 = t0 * t0; a1 = t1 * t1; }
                else             { a0 = p0 * t0; a1 = p1 * t1; }
                v2f a = {a0, a1};
                v2f b = {Tf[c][0], Tf[c][1]};
                d = wmma_16x16x4_f32(a, b, d);       // Sx = M_p * T
            }
            asm volatile("" ::: "memory");           // order store->reload (DS in-order)
#pragma unroll
            for (int r = 0; r < 8; ++r)
                smem[4 * TILE + (r + 8 * half) * LST + l16] = d[r];
            asm volatile("" ::: "memory");
            v8f d2 = (v8f){0,0,0,0,0,0,0,0};
#pragma unroll
            for (int c = 0; c < 4; ++c) {
                v2f a = {Tf[c][0], Tf[c][1]};        // A' = T (symmetric)
                int kg = 4 * c + 2 * half;
                v2f b = {smem[4 * TILE + (kg + 0) * LST + l16],
                         smem[4 * TILE + (kg + 1) * LST + l16]};
                d2 = wmma_16x16x4_f32(a, b, d2);     // Sxy = T * Sx
            }
            Pnew[p] = d2;
        }

        // ---- rotate z ring; emit output plane zo = zp-1 ----
#pragma unroll
        for (int p = 0; p < 5; ++p) { Pa[p] = Pb[p]; Pb[p] = Pc[p]; Pc[p] = Pnew[p]; }

        if (zp >= z0 + 1) {
            v8f W[5];
#pragma unroll
            for (int p = 0; p < 5; ++p) W[p] = Pa[p] + Pb[p] + Pc[p];
#pragma unroll
            for (int r = 0; r < 8; ++r) {
                int yl = r + 8 * half, xl = l16;
                int gx = x0 + xl, gy = y0 + yl;
                bool valid = (xl >= lox) && (xl <= hix) &&
                             (yl >= loy) && (yl <= hiy) &&
                             (gx < N) && (gy < N);
                if (valid) {
                    float Is  = W[0][r], Js  = W[1][r];
                    float IIs = W[2][r], JJs = W[3][r], IJs = W[4][r];
                    const float inv_kv = 1.0f / 27.0f;
                    float uI = Is * inv_kv, uJ = Js * inv_kv;
                    float cross = IJs - uJ * Is;
                    float Ivar  = IIs - uI * Is;
                    float Jvar  = JJs - uJ * Js;
                    acc += (cross * cross) / (Ivar * Jvar + 1e-5f);
                }
            }
        }
    }

    // wave32 reduction, one partial per block
#pragma unroll
    for (int off = 16; off > 0; off >>= 1) acc += __shfl_xor(acc, off, 32);
    if (lane == 0) partial[bid] = acc;
}

__global__ __launch_bounds__(256)
void lncc_finalize(const float* __restrict__ partial, int nPart,
                   float* __restrict__ out)
{
    __shared__ float red[256];
    float s = 0.0f;
    for (int i = threadIdx.x; i < nPart; i += 256) s += partial[i];
    red[threadIdx.x] = s;
    __syncthreads();
    for (int w = 128; w > 0; w >>= 1) {
        if ((int)threadIdx.x < w) red[threadIdx.x] += red[threadIdx.x + w];
        __syncthreads();
    }
    if (threadIdx.x == 0)
        out[0] = -red[0] / (float)((size_t)N * N * N);
}

extern "C" void kernel_launch(void* const* d_in, const int* in_sizes, int n_in,
                              void* d_out, int out_size, void* d_ws, size_t ws_size,
                              hipStream_t stream) {
    const float* pred = (const float*)d_in[0];
    const float* targ = (const float*)d_in[1];
    float* out     = (float*)d_out;
    float* partial = (float*)d_ws;   // NBLK floats (6.3 KB)

    lncc_main<<<NBLK, 32, 0, stream>>>(pred, targ, partial);
    lncc_finalize<<<1, 256, 0, stream>>>(partial, NBLK, out);
}